// HookedMambaLayer_74929999446531
// MI455X (gfx1250) — compile-verified
//
#include <hip/hip_runtime.h>
#include <math.h>

// ---------------------------------------------------------------------------
// Mamba layer, MI455X (gfx1250, wave32). GEMMs on the native fp32 matrix pipe
// (V_WMMA_F32_16X16X4_F32); global->LDS staging through the CDNA5 async path
// (GLOBAL_LOAD_ASYNC_TO_LDS_B128 + s_wait_asynccnt) when the toolchain
// exposes it, with a sync ds_store fallback.
// ---------------------------------------------------------------------------

typedef float v2f __attribute__((ext_vector_type(2)));
typedef float v8f __attribute__((ext_vector_type(8)));

// Types matching the async-LDS builtin's signature (from hipcc diagnostics):
//   arg0: int __vector(4) addrspace(1)*   arg1: int __vector(4) addrspace(3)*
typedef int v4i_gcc __attribute__((vector_size(16)));
typedef __attribute__((address_space(1))) v4i_gcc* as1_v4i_p;
typedef __attribute__((address_space(3))) v4i_gcc* as3_v4i_p;

#define LSEQ   2048
#define DMODEL 1024
#define EDIM   2048
#define NSTATE 16
#define DDELTA 64

#if defined(__has_builtin)
#if __has_builtin(__builtin_amdgcn_global_load_async_to_lds_b128)
#define USE_ASYNC_LDS 1
#endif
#endif

__device__ __forceinline__ float sigmoid_f(float x) {
  return 1.0f / (1.0f + __expf(-x));
}

__device__ __forceinline__ void wait_async0() {
#if defined(__AMDGCN__)
#if defined(__has_builtin) && __has_builtin(__builtin_amdgcn_s_wait_asynccnt)
  __builtin_amdgcn_s_wait_asynccnt(0);
#else
  asm volatile("s_wait_asynccnt 0x0" ::: "memory");
#endif
#endif
}

// ---------------- RMSNorm: x = resid * rsqrt(mean(resid^2)+eps) * norm_w ----
__global__ __launch_bounds__(256) void rmsnorm_kernel(
    const float* __restrict__ resid, const float* __restrict__ norm_w,
    float* __restrict__ xout) {
  const int l = blockIdx.x;
  const int tid = threadIdx.x;
  __shared__ float red[8];
  const float* row = resid + (size_t)l * DMODEL;
  float s = 0.f;
  for (int d = tid; d < DMODEL; d += 256) { float v = row[d]; s += v * v; }
  #pragma unroll
  for (int off = 16; off > 0; off >>= 1) s += __shfl_down(s, off, 32);
  if ((tid & 31) == 0) red[tid >> 5] = s;
  __syncthreads();
  if (tid == 0) {
    float t = 0.f;
    #pragma unroll
    for (int i = 0; i < 8; ++i) t += red[i];
    red[0] = rsqrtf(t / (float)DMODEL + 1e-5f);
  }
  __syncthreads();
  const float r = red[0];
  float* orow = xout + (size_t)l * DMODEL;
  for (int d = tid; d < DMODEL; d += 256) orow[d] = row[d] * r * norm_w[d];
}

// ---------------- Generic WMMA GEMM (NT): C[M,N] = act(A[M,K] @ W[N,K]^T) ---
// Block tile 128x64, 8 waves in a 4(M) x 2(N) grid, each wave owns a 2x2
// arrangement of 16x16 D tiles (32x32 per wave): 4 LDS fragment loads feed
// 4 WMMAs with full register reuse of A/B fragments. K staged in LDS chunks
// of 32 (padded row stride 36: conflict-free, every row 16B aligned).
// f32 A-frag layout (ISA 7.12.2): lane m = lane&15; the VGPR pair holds
// K = {0,1} (lanes 0-15) / {2,3} (lanes 16-31) of each k-quad.
template <int ACT, bool HAS_BIAS, bool HAS_ADD>
__global__ __launch_bounds__(256) void wmma_gemm_nt(
    const float* __restrict__ A, const float* __restrict__ W,
    float* __restrict__ C, const float* __restrict__ bias,
    const float* __restrict__ addend, int M, int N, int K) {
  __shared__ float As[128][36];
  __shared__ float Ws[64][36];
  const int tid   = threadIdx.x;
  const int lane  = tid & 31;
  const int wave  = tid >> 5;
  const int laneN = lane & 15;
  const int khalf = (lane >> 4) << 1;  // 0 or 2
  const int rowH  = (lane >> 4) * 8;   // C/D layout: lanes 16-31 hold M = r+8
  const int m0 = blockIdx.y * 128;
  const int n0 = blockIdx.x * 64;
  const int wm = (wave >> 1) * 32;     // 0..96  (block-relative M offset)
  const int wn = (wave & 1) * 32;      // 0 / 32 (block-relative N offset)

  // loader mapping: A chunk 128x32 (16 floats/thread), W chunk 64x32 (8/thread)
  const int arow = tid >> 1;           // 0..127
  const int acol = (tid & 1) * 16;     // 0 or 16
  const int wrow = tid >> 2;           // 0..63
  const int wcol = (tid & 3) * 8;      // 0,8,16,24
  // OOB weight rows (N not a multiple of 64): clamp the *global* source row to
  // a valid one; the tiles consuming those LDS rows are never stored.
  const int gwrow = (n0 + wrow < N) ? (n0 + wrow) : 0;

  v8f acc00 = {0.f, 0.f, 0.f, 0.f, 0.f, 0.f, 0.f, 0.f};
  v8f acc01 = acc00, acc10 = acc00, acc11 = acc00;

  for (int kk = 0; kk < K; kk += 32) {
#if defined(USE_ASYNC_LDS)
    __syncthreads();  // previous iteration's LDS reads done before overwrite
    #pragma unroll
    for (int i = 0; i < 4; ++i) {
      __builtin_amdgcn_global_load_async_to_lds_b128(
          (as1_v4i_p)(A + (size_t)(m0 + arow) * K + kk + acol + i * 4),
          (as3_v4i_p)&As[arow][acol + i * 4], 0, 0);
    }
    #pragma unroll
    for (int i = 0; i < 2; ++i) {
      __builtin_amdgcn_global_load_async_to_lds_b128(
          (as1_v4i_p)(W + (size_t)gwrow * K + kk + wcol + i * 4),
          (as3_v4i_p)&Ws[wrow][wcol + i * 4], 0, 0);
    }
    wait_async0();
    __syncthreads();
#else
    float4 a4[4];
    #pragma unroll
    for (int i = 0; i < 4; ++i)
      a4[i] = *(const float4*)(A + (size_t)(m0 + arow) * K + kk + acol + i * 4);
    const float4 w4a = *(const float4*)(W + (size_t)gwrow * K + kk + wcol);
    const float4 w4b = *(const float4*)(W + (size_t)gwrow * K + kk + wcol + 4);
    __syncthreads();
    #pragma unroll
    for (int i = 0; i < 4; ++i) *(float4*)&As[arow][acol + i * 4] = a4[i];
    *(float4*)&Ws[wrow][wcol]     = w4a;
    *(float4*)&Ws[wrow][wcol + 4] = w4b;
    __syncthreads();
#endif
    #pragma unroll
    for (int k = 0; k < 32; k += 4) {
      const v2f a0 = *(const v2f*)&As[wm + laneN][k + khalf];
      const v2f a1 = *(const v2f*)&As[wm + 16 + laneN][k + khalf];
      const v2f b0 = *(const v2f*)&Ws[wn + laneN][k + khalf];
      const v2f b1 = *(const v2f*)&Ws[wn + 16 + laneN][k + khalf];
      acc00 = __builtin_amdgcn_wmma_f32_16x16x4_f32(false, a0, false, b0,
                                                    (short)0, acc00, false, false);
      acc01 = __builtin_amdgcn_wmma_f32_16x16x4_f32(false, a0, false, b1,
                                                    (short)0, acc01, false, false);
      acc10 = __builtin_amdgcn_wmma_f32_16x16x4_f32(false, a1, false, b0,
                                                    (short)0, acc10, false, false);
      acc11 = __builtin_amdgcn_wmma_f32_16x16x4_f32(false, a1, false, b1,
                                                    (short)0, acc11, false, false);
    }
  }

  auto store_tile = [&](const v8f& acc, int mbase, int nbase) {
    const int col = n0 + wn + nbase + laneN;
    #pragma unroll
    for (int r = 0; r < 8; ++r) {
      const int row = m0 + wm + mbase + r + rowH;
      if (row < M && col < N) {
        float v = acc[r];
        if (HAS_BIAS) v += bias[col];
        if (ACT == 1) v = (v > 20.f) ? v : __logf(1.f + __expf(v));  // softplus
        if (HAS_ADD) v += addend[(size_t)row * N + col];
        C[(size_t)row * N + col] = v;
      }
    }
  };
  store_tile(acc00, 0, 0);
  store_tile(acc01, 0, 16);
  store_tile(acc10, 16, 0);
  store_tile(acc11, 16, 16);
}

// ---------------- Causal depthwise conv (width 4) + SiLU --------------------
__global__ __launch_bounds__(256) void conv_silu_kernel(
    const float* __restrict__ xp, const float* __restrict__ cw,
    const float* __restrict__ cb, float* __restrict__ xs) {
  const int idx = blockIdx.x * 256 + threadIdx.x;   // over L*E
  const int e = idx & (EDIM - 1);
  const int l = idx >> 11;                          // E = 2^11
  const float4 w4 = *(const float4*)(cw + (size_t)e * 4);
  const float wj[4] = {w4.x, w4.y, w4.z, w4.w};
  float acc = cb[e];
  #pragma unroll
  for (int j = 0; j < 4; ++j) {
    const int ls = l - 3 + j;
    if (ls >= 0) acc += xp[(size_t)ls * EDIM + e] * wj[j];
  }
  xs[idx] = acc * sigmoid_f(acc);
}

// ---------------- Sequential SSM scan, fused D-skip + SiLU-gate epilogue ----
// One thread per channel e; the N=16 state vector stays in registers. Serial
// in L (inherent recurrence), fully parallel across E = 2048 lanes.
__global__ __launch_bounds__(256) void ssm_scan_kernel(
    const float* __restrict__ delta, const float* __restrict__ xs,
    const float* __restrict__ Bm, const float* __restrict__ Cm,
    const float* __restrict__ A_log, const float* __restrict__ w_D,
    const float* __restrict__ skip, float* __restrict__ ymul) {
  const int e = blockIdx.x * 256 + threadIdx.x;  // 0..E-1
  float Arow[NSTATE], h[NSTATE];
  #pragma unroll
  for (int n = 0; n < NSTATE; ++n) {
    Arow[n] = -__expf(A_log[(size_t)e * NSTATE + n]);
    h[n] = 0.f;
  }
  const float wd = w_D[e];
  for (int l = 0; l < LSEQ; ++l) {
    const float d  = delta[(size_t)l * EDIM + e];
    const float xv = xs[(size_t)l * EDIM + e];
    const float dx = d * xv;
    const float4* bp = (const float4*)(Bm + (size_t)l * NSTATE);
    const float4* cp = (const float4*)(Cm + (size_t)l * NSTATE);
    float y = 0.f;
    #pragma unroll
    for (int q = 0; q < 4; ++q) {
      const float4 b4 = bp[q];
      const float4 c4 = cp[q];
      const float bb[4] = {b4.x, b4.y, b4.z, b4.w};
      const float cc[4] = {c4.x, c4.y, c4.z, c4.w};
      #pragma unroll
      for (int j = 0; j < 4; ++j) {
        const int n = q * 4 + j;
        const float ab = __expf(d * Arow[n]);   // exp(delta*A) < 1
        h[n] = ab * h[n] + dx * bb[j];
        y += h[n] * cc[j];
      }
    }
    const float sk = skip[(size_t)l * EDIM + e];
    ymul[(size_t)l * EDIM + e] = (y + xv * wd) * (sk * sigmoid_f(sk));
  }
}

// ---------------------------------------------------------------------------
extern "C" void kernel_launch(void* const* d_in, const int* in_sizes, int n_in,
                              void* d_out, int out_size, void* d_ws,
                              size_t ws_size, hipStream_t stream) {
  (void)in_sizes; (void)n_in; (void)out_size; (void)ws_size;
  const float* resid  = (const float*)d_in[0];
  const float* norm_w = (const float*)d_in[1];
  const float* skip_w = (const float*)d_in[2];
  const float* in_w   = (const float*)d_in[3];
  const float* conv_w = (const float*)d_in[4];
  const float* conv_b = (const float*)d_in[5];
  const float* wd1    = (const float*)d_in[6];
  const float* wd2    = (const float*)d_in[7];
  const float* wd2_b  = (const float*)d_in[8];
  const float* w_B    = (const float*)d_in[9];
  const float* w_C    = (const float*)d_in[10];
  const float* A_log  = (const float*)d_in[11];
  const float* w_D    = (const float*)d_in[12];
  const float* out_w  = (const float*)d_in[13];
  float* out = (float*)d_out;

  // scratch layout (~93 MB of f32)
  float* ws    = (float*)d_ws;
  float* x     = ws; ws += (size_t)LSEQ * DMODEL;
  float* skip  = ws; ws += (size_t)LSEQ * EDIM;
  float* xp    = ws; ws += (size_t)LSEQ * EDIM;
  float* xs    = ws; ws += (size_t)LSEQ * EDIM;
  float* t1    = ws; ws += (size_t)LSEQ * DDELTA;
  float* delta = ws; ws += (size_t)LSEQ * EDIM;
  float* Bm    = ws; ws += (size_t)LSEQ * NSTATE;
  float* Cm    = ws; ws += (size_t)LSEQ * NSTATE;
  float* ymul  = ws;

  const dim3 blk(256);

  rmsnorm_kernel<<<LSEQ, blk, 0, stream>>>(resid, norm_w, x);

  // skip = x @ skip_w^T ; xp = x @ in_w^T      (M=2048, N=2048, K=1024)
  wmma_gemm_nt<0, false, false><<<dim3(EDIM / 64, LSEQ / 128), blk, 0, stream>>>(
      x, skip_w, skip, nullptr, nullptr, LSEQ, EDIM, DMODEL);
  wmma_gemm_nt<0, false, false><<<dim3(EDIM / 64, LSEQ / 128), blk, 0, stream>>>(
      x, in_w, xp, nullptr, nullptr, LSEQ, EDIM, DMODEL);

  conv_silu_kernel<<<(LSEQ * EDIM) / 256, blk, 0, stream>>>(xp, conv_w, conv_b, xs);

  // t1 = xs @ wd1^T                            (M=2048, N=64, K=2048)
  wmma_gemm_nt<0, false, false><<<dim3(1, LSEQ / 128), blk, 0, stream>>>(
      xs, wd1, t1, nullptr, nullptr, LSEQ, DDELTA, EDIM);
  // delta = softplus(t1 @ wd2^T + wd2_b)       (M=2048, N=2048, K=64)
  wmma_gemm_nt<1, true, false><<<dim3(EDIM / 64, LSEQ / 128), blk, 0, stream>>>(
      t1, wd2, delta, wd2_b, nullptr, LSEQ, EDIM, DDELTA);
  // Bm / Cm = xs @ w_{B,C}^T                   (M=2048, N=16, K=2048)
  wmma_gemm_nt<0, false, false><<<dim3(1, LSEQ / 128), blk, 0, stream>>>(
      xs, w_B, Bm, nullptr, nullptr, LSEQ, NSTATE, EDIM);
  wmma_gemm_nt<0, false, false><<<dim3(1, LSEQ / 128), blk, 0, stream>>>(
      xs, w_C, Cm, nullptr, nullptr, LSEQ, NSTATE, EDIM);

  ssm_scan_kernel<<<EDIM / 256, blk, 0, stream>>>(delta, xs, Bm, Cm, A_log,
                                                  w_D, skip, ymul);

  // out = resid + ymul @ out_w^T               (M=2048, N=1024, K=2048)
  wmma_gemm_nt<0, false, true><<<dim3(DMODEL / 64, LSEQ / 128), blk, 0, stream>>>(
      ymul, out_w, out, nullptr, resid, LSEQ, DMODEL, EDIM);
}